// StepwiseMonotonicAttention_73727408603343
// MI455X (gfx1250) — compile-verified
//
#include <hip/hip_runtime.h>
#include <math.h>

typedef __attribute__((ext_vector_type(2))) float v2f;
typedef __attribute__((ext_vector_type(8))) float v8f;

#define BB   64
#define TT   1024
#define DQ   1024
#define DV   512
#define AA   128
#define TCH  8            // number of T-chunks for context partials
#define TPC  (TT / TCH)   // 128 timesteps per chunk
#define SIGMOID_NOISE 2.0f

// ---------------------------------------------------------------------------
// Kernel 0: w_v = v_g * v_dir / ||v_dir||   (128 elems, one block)
// ---------------------------------------------------------------------------
__global__ __launch_bounds__(128)
void wv_kernel(const float* __restrict__ v_dir, const float* __restrict__ v_g,
               float* __restrict__ w_v) {
    __shared__ float red[128];
    const int i = threadIdx.x;
    const float d = v_dir[i];
    red[i] = d * d;
    __syncthreads();
    for (int s = 64; s > 0; s >>= 1) {
        if (i < s) red[i] += red[i + s];
        __syncthreads();
    }
    const float inv_norm = rsqrtf(red[0]);
    w_v[i] = v_g[0] * d * inv_norm;
}

// ---------------------------------------------------------------------------
// Kernel 1: pq = query @ Wq^T  via V_WMMA_F32_16X16X4_F32
//   One wave (32 threads) per 16x16 tile of pq; grid (64/16, 128/16) = (4, 8).
//   A (16x4 f32): lanes 0-15 hold K=0,1 in vgpr pair; lanes 16-31 hold K=2,3.
//   B (4x16 f32): same K striping, N = lane%16.
//   D (16x16 f32): vgpr r holds row m0 + r + 8*(lane/16), col n0 + lane%16.
// ---------------------------------------------------------------------------
__global__ __launch_bounds__(32)
void pq_wmma_kernel(const float* __restrict__ query,   // [64, 1024]
                    const float* __restrict__ Wq,      // [128, 1024]
                    float* __restrict__ pq) {          // [64, 128]
    const int m0   = blockIdx.x * 16;
    const int n0   = blockIdx.y * 16;
    const int lane = threadIdx.x;     // 0..31, wave32
    const int lo   = lane & 15;
    const int hi   = lane >> 4;       // 0 or 1

    const float* __restrict__ qrow = query + (size_t)(m0 + lo) * DQ + 2 * hi;
    const float* __restrict__ wrow = Wq    + (size_t)(n0 + lo) * DQ + 2 * hi;

    v8f acc = {};
    for (int k = 0; k < DQ; k += 4) {
        const v2f a = *(const v2f*)(qrow + k);   // A[M=lo, K=k+2*hi .. +1]
        const v2f b = *(const v2f*)(wrow + k);   // B[K=k+2*hi .., N=lo]
        acc = __builtin_amdgcn_wmma_f32_16x16x4_f32(
            /*neg_a=*/false, a, /*neg_b=*/false, b,
            /*c_mod=*/(short)0, acc, /*reuse_a=*/false, /*reuse_b=*/false);
    }
#pragma unroll
    for (int r = 0; r < 8; ++r) {
        pq[(size_t)(m0 + r + 8 * hi) * AA + (n0 + lo)] = acc[r];
    }
}

// ---------------------------------------------------------------------------
// Kernel 2: p[b,t] = sigmoid(score_bias + v_bias + 2*noise
//                            + sum_a tanh(pq[b,a] + pm[b,t,a]) * w_v[a])
//   One wave per (b,t); each lane handles a contiguous float4 of A.
// ---------------------------------------------------------------------------
__global__ __launch_bounds__(256)
void score_kernel(const float* __restrict__ pm,     // [64, 1024, 128]
                  const float* __restrict__ pq,     // [64, 128]
                  const float* __restrict__ w_v,    // [128]
                  const float* __restrict__ noise,  // [64, 1024]
                  const float* __restrict__ score_bias,
                  const float* __restrict__ v_bias,
                  float* __restrict__ p_out) {      // [64, 1024]
    const int wave = (int)((blockIdx.x * blockDim.x + threadIdx.x) >> 5);
    const int lane = threadIdx.x & 31;
    if (wave >= BB * TT) return;
    const int b = wave >> 10;            // wave / T

    const float4 pm4 = *(const float4*)(pm + (size_t)wave * AA + lane * 4);
    const float4 pq4 = *(const float4*)(pq + (size_t)b * AA + lane * 4);
    const float4 wv4 = *(const float4*)(w_v + lane * 4);

    float s = tanhf(pm4.x + pq4.x) * wv4.x
            + tanhf(pm4.y + pq4.y) * wv4.y
            + tanhf(pm4.z + pq4.z) * wv4.z
            + tanhf(pm4.w + pq4.w) * wv4.w;

    // wave32 butterfly reduction
    for (int off = 16; off > 0; off >>= 1) s += __shfl_xor(s, off, 32);

    if (lane == 0) {
        const float sc = score_bias[0] + s + v_bias[0]
                       + SIGMOID_NOISE * noise[wave];
        p_out[wave] = 1.0f / (1.0f + expf(-sc));
    }
}

// ---------------------------------------------------------------------------
// Kernel 3: alignment = prev*p + shift_right(prev*(1-p)); masked -> 0
// ---------------------------------------------------------------------------
__global__ __launch_bounds__(256)
void align_kernel(const float* __restrict__ p,
                  const float* __restrict__ prev,
                  const unsigned char* __restrict__ mask,
                  float* __restrict__ align_out) {
    const int i = (int)(blockIdx.x * blockDim.x + threadIdx.x);
    if (i >= BB * TT) return;
    const int t = i & (TT - 1);
    float al = prev[i] * p[i];
    if (t > 0) al += prev[i - 1] * (1.0f - p[i - 1]);
    align_out[i] = mask[i] ? 0.0f : al;
}

// ---------------------------------------------------------------------------
// Kernel 4: context partials over T-chunks.
//   grid (B, TCH); 128 threads, each owns a float4 of Dv (coalesced rows).
// ---------------------------------------------------------------------------
__global__ __launch_bounds__(128)
void ctx_partial_kernel(const float* __restrict__ align,   // [64, 1024]
                        const float* __restrict__ memory,  // [64, 1024, 512]
                        float* __restrict__ partial) {     // [TCH, 64, 512]
    const int b  = blockIdx.x;
    const int c  = blockIdx.y;
    const int d4 = threadIdx.x;   // 0..127 -> d = 4*d4

    const float* __restrict__ mrow =
        memory + ((size_t)b * TT + (size_t)c * TPC) * DV + d4 * 4;
    const float* __restrict__ arow = align + (size_t)b * TT + (size_t)c * TPC;

    float4 acc = {0.f, 0.f, 0.f, 0.f};
    for (int t = 0; t < TPC; ++t) {
        const float  a = arow[t];
        const float4 m = *(const float4*)(mrow + (size_t)t * DV);
        acc.x += a * m.x;
        acc.y += a * m.y;
        acc.z += a * m.z;
        acc.w += a * m.w;
    }
    *(float4*)(partial + ((size_t)c * BB + b) * DV + d4 * 4) = acc;
}

// ---------------------------------------------------------------------------
// Kernel 5: deterministic reduction of the TCH partials -> context
// ---------------------------------------------------------------------------
__global__ __launch_bounds__(256)
void ctx_reduce_kernel(const float* __restrict__ partial,  // [TCH, 64, 512]
                       float* __restrict__ ctx) {          // [64, 512]
    const int i = (int)(blockIdx.x * blockDim.x + threadIdx.x);
    if (i >= BB * DV) return;
    float s = 0.f;
#pragma unroll
    for (int c = 0; c < TCH; ++c) s += partial[(size_t)c * BB * DV + i];
    ctx[i] = s;
}

// ---------------------------------------------------------------------------
extern "C" void kernel_launch(void* const* d_in, const int* in_sizes, int n_in,
                              void* d_out, int out_size, void* d_ws, size_t ws_size,
                              hipStream_t stream) {
    (void)in_sizes; (void)n_in; (void)out_size; (void)ws_size;

    const float*         query  = (const float*)d_in[0];           // [64,1024]
    const float*         memory = (const float*)d_in[1];           // [64,1024,512]
    const float*         pm     = (const float*)d_in[2];           // [64,1024,128]
    const float*         prev   = (const float*)d_in[3];           // [64,1024]
    const unsigned char* mask   = (const unsigned char*)d_in[4];   // [64,1024] bool
    const float*         noise  = (const float*)d_in[5];           // [64,1024]
    const float*         Wq     = (const float*)d_in[6];           // [128,1024]
    const float*         v_dir  = (const float*)d_in[7];           // [128]
    const float*         v_g    = (const float*)d_in[8];           // scalar
    const float*         v_bias = (const float*)d_in[9];           // scalar
    const float*         sbias  = (const float*)d_in[10];          // scalar

    float* out_ctx   = (float*)d_out;                  // [64, 512]
    float* out_align = (float*)d_out + BB * DV;        // [64, 1024]

    // workspace carve-up (floats)
    float* ws      = (float*)d_ws;
    float* w_v     = ws;                               // 128
    float* pq      = w_v + AA;                         // 64*128   = 8192
    float* p       = pq + BB * AA;                     // 64*1024  = 65536
    float* partial = p + BB * TT;                      // 8*64*512 = 262144

    // 0) weight-norm v
    wv_kernel<<<1, 128, 0, stream>>>(v_dir, v_g, w_v);

    // 1) pq = query @ Wq^T via f32 WMMA (one wave per 16x16 tile)
    dim3 pq_grid(BB / 16, AA / 16);
    pq_wmma_kernel<<<pq_grid, 32, 0, stream>>>(query, Wq, pq);

    // 2) p = sigmoid(score)  (one wave per (b,t), 8 waves per block)
    const int n_bt = BB * TT;
    score_kernel<<<(n_bt * 32 + 255) / 256, 256, 0, stream>>>(
        pm, pq, w_v, noise, sbias, v_bias, p);

    // 3) alignment (also the second output)
    align_kernel<<<(n_bt + 255) / 256, 256, 0, stream>>>(p, prev, mask, out_align);

    // 4) context partials over T-chunks, then deterministic reduce
    dim3 ctx_grid(BB, TCH);
    ctx_partial_kernel<<<ctx_grid, 128, 0, stream>>>(out_align, memory, partial);
    ctx_reduce_kernel<<<(BB * DV + 255) / 256, 256, 0, stream>>>(partial, out_ctx);
}